// jslab_kt_78554951844273
// MI455X (gfx1250) — compile-verified
//
#include <hip/hip_runtime.h>

// ---------------------------------------------------------------------------
// Slab-ocean linear recurrence  z[n+1] = lam*z[n] + b[n]  (z = U + iV)
//   lam  = (1 - DT*k1) - i*DT*fc       (constant: Kt rows are all identical)
//   b[n] = DT*k0*(tax[n] + i*tay[n])   (linearly interpolated hourly forcing)
// Chunk-parallel scan; intra-chunk prefix materialization via f32 WMMA
// (T[i,j] = lam^(i-j) lower-triangular Toeplitz, complex -> two real GEMMs).
// ---------------------------------------------------------------------------

#define NSTEPS 525600
#define NTF    8760
#define NSUB   60
#define FDT    60.0f
#define NDT    36
#define SEG    1024
#define NBLK   ((NSTEPS + SEG - 1) / SEG)   // 514

typedef __attribute__((ext_vector_type(2))) float v2f;
typedef __attribute__((ext_vector_type(8))) float v8f;

// ---- shared scalar setup ---------------------------------------------------
__device__ __forceinline__ void slab_consts(const float* pk, float fc,
                                            float& a, float& c, float& dtk0,
                                            float& r, float& th) {
    float k0 = 0.f, k1 = 0.f;
    for (int j = 0; j < NDT; ++j) {
        k0 += expf(pk[2 * j]);
        k1 += expf(pk[2 * j + 1]);
    }
    k0 *= (1.0f / NDT);
    k1 *= (1.0f / NDT);
    a    = 1.0f - FDT * k1;       // Re(lam)
    c    = FDT * fc;              // -Im(lam): lam = a - i*c
    dtk0 = FDT * k0;
    r    = sqrtf(a * a + c * c);  // |lam|
    th   = atan2f(-c, a);         // arg(lam)
}

__device__ __forceinline__ void slab_forcing(int it, const float* __restrict__ TAx,
                                             const float* __restrict__ TAy,
                                             float dtk0, float& u, float& v) {
    if (it >= NSTEPS) { u = 0.f; v = 0.f; return; }  // pad tail of last block
    int   itf = it / NSUB;
    float aa  = (float)(it - itf * NSUB) * (1.0f / NSUB);
    int   sup = itf + 1;
    if (sup >= NTF) sup = NTF - 1;                   // jnp wrap of index -1
    float om = 1.0f - aa;
    u = dtk0 * (om * TAx[itf] + aa * TAx[sup]);
    v = dtk0 * (om * TAy[itf] + aa * TAy[sup]);
}

// lam^n via polar form (accurate; far powers underflow to 0, which is exact
// in the limit since |lam| < 1)
__device__ __forceinline__ void lam_pow(float r, float th, float n,
                                        float& pr, float& pi) {
    float mag = expf(n * logf(r));
    float ang = n * th;
    pr = mag * cosf(ang);
    pi = mag * sinf(ang);
}

// ---- Kernel 1: per-block weighted sums  S_blk = sum lam^(SEG-1-m) b[m] -----
__global__ void k1_blocksums(const float* __restrict__ pk,
                             const float* __restrict__ TAx,
                             const float* __restrict__ TAy,
                             const float* __restrict__ fcp,
                             float* __restrict__ S) {
    __shared__ float sc[5];
    __shared__ float redr[256], redi[256];
    const int tid = threadIdx.x, blk = blockIdx.x;

    if (tid == 0) {
        float a, c, dtk0, r, th;
        slab_consts(pk, fcp[0], a, c, dtk0, r, th);
        sc[0] = a; sc[1] = c; sc[2] = dtk0; sc[3] = r; sc[4] = th;
    }
    __syncthreads();
    const float a = sc[0], c = sc[1], dtk0 = sc[2], rr = sc[3], th = sc[4];

    // 4 serial steps per thread (local Horner in lam)
    const int base = blk * SEG + tid * 4;
    float wr = 0.f, wi = 0.f;
#pragma unroll
    for (int j = 0; j < 4; ++j) {
        float u, v;
        slab_forcing(base + j, TAx, TAy, dtk0, u, v);
        float nr = a * wr + c * wi + u;   // lam*w + b,  lam = (a, -c)
        float ni = a * wi - c * wr + v;
        wr = nr; wi = ni;
    }
    // scale by lam^(4*(255-tid)) so exponents line up with SEG-1-m
    float pr, pi;
    lam_pow(rr, th, (float)(4 * (255 - tid)), pr, pi);
    redr[tid] = wr * pr - wi * pi;
    redi[tid] = wr * pi + wi * pr;
    __syncthreads();
    for (int s = 128; s > 0; s >>= 1) {
        if (tid < s) { redr[tid] += redr[tid + s]; redi[tid] += redi[tid + s]; }
        __syncthreads();
    }
    if (tid == 0) { S[2 * blk] = redr[0]; S[2 * blk + 1] = redi[0]; }
}

// ---- Kernel 2: scan of 514 block sums -> block-start states Z --------------
__global__ void k2_scan(const float* __restrict__ pk,
                        const float* __restrict__ fcp,
                        const float* __restrict__ S,
                        float* __restrict__ Z) {
    if (threadIdx.x != 0) return;
    float a, c, dtk0, r, th;
    slab_consts(pk, fcp[0], a, c, dtk0, r, th);
    float lr, li;
    lam_pow(r, th, (float)SEG, lr, li);   // lam^1024
    float zr = 0.f, zi = 0.f;
    Z[0] = 0.f; Z[1] = 0.f;
    for (int b = 1; b < NBLK; ++b) {
        float nr = lr * zr - li * zi + S[2 * (b - 1)];
        float ni = lr * zi + li * zr + S[2 * (b - 1) + 1];
        zr = nr; zi = ni;
        Z[2 * b] = zr; Z[2 * b + 1] = zi;
    }
}

// ---- Kernel 3: materialize all prefixes with WMMA --------------------------
// Per wave: 8 complex chunks of 16 steps -> one 16x16 tile.
//   D = P @ [u v ...] + Q @ [-v u ...]   (P=Re(lam^(i-j)), Q=Im, lower-tri)
// Zero-padded LDS power table -> unconditional ds_loads, all operand
// construction hoisted so the 8 v_wmma issue back-to-back.
__global__ void k3_materialize(const float* __restrict__ pk,
                               const float* __restrict__ TAx,
                               const float* __restrict__ TAy,
                               const float* __restrict__ fcp,
                               const float* __restrict__ Z,
                               float* __restrict__ out) {
    // T[d+16] = lam^d for d in [0,16], 0 for d in [-16,-1]
    __shared__ float s_Tre[33], s_Tim[33];
    __shared__ float s_p16r[8], s_p16i[8];   // lam^(16k), k=0..7
    __shared__ float s_p128r, s_p128i;       // lam^128
    __shared__ float s_wr[8], s_wi[8];       // per-wave totals
    __shared__ float s_dtk0;

    const int tid = threadIdx.x, blk = blockIdx.x;
    if (tid < 16) { s_Tre[tid] = 0.f; s_Tim[tid] = 0.f; }  // negative-d pad
    if (tid == 0) {
        float a, c, dtk0, r, th;
        slab_consts(pk, fcp[0], a, c, dtk0, r, th);
        s_dtk0 = dtk0;
        s_Tre[16] = 1.f; s_Tim[16] = 0.f;
        for (int d = 1; d <= 16; ++d) {      // lam^d = lam * lam^(d-1)
            s_Tre[16 + d] = a * s_Tre[15 + d] + c * s_Tim[15 + d];
            s_Tim[16 + d] = a * s_Tim[15 + d] - c * s_Tre[15 + d];
        }
        float g = s_Tre[32], h = s_Tim[32];  // lam^16
        s_p16r[0] = 1.f; s_p16i[0] = 0.f;
        for (int k = 1; k < 8; ++k) {
            s_p16r[k] = s_p16r[k - 1] * g - s_p16i[k - 1] * h;
            s_p16i[k] = s_p16r[k - 1] * h + s_p16i[k - 1] * g;
        }
        s_p128r = s_p16r[7] * g - s_p16i[7] * h;
        s_p128i = s_p16r[7] * h + s_p16i[7] * g;
    }
    __builtin_prefetch(TAx + (blk * SEG) / NSUB, 0, 0);   // global_prefetch_b8
    __builtin_prefetch(TAy + (blk * SEG) / NSUB, 0, 0);
    __syncthreads();

    const int   w    = tid >> 5;          // wave 0..7, 128 steps each
    const int   lane = tid & 31;
    const int   hi   = lane >> 4;
    const int   n    = lane & 15;         // tile column
    const int   q    = n >> 1;            // chunk 0..7 within wave
    const int   comp = n & 1;             // 0 = Re column, 1 = Im column
    const float dtk0 = s_dtk0;
    const int   cbase = blk * SEG + w * 128 + q * 16;  // lane's chunk start

    // -- hoisted operand construction (no EXEC games, unconditional ds loads)
    v2f AP[4], AQ[4], BP[4], BQ[4];
#pragma unroll
    for (int kb = 0; kb < 4; ++kb) {
        int i0 = n - (4 * kb + 2 * hi) + 16;   // in [0,31]
        AP[kb] = v2f{ s_Tre[i0], s_Tre[i0 - 1] };
        AQ[kb] = v2f{ s_Tim[i0], s_Tim[i0 - 1] };
    }
    float bu[8], bv[8];
#pragma unroll
    for (int kb = 0; kb < 4; ++kb) {
        int j0 = 4 * kb + 2 * hi;
        slab_forcing(cbase + j0,     TAx, TAy, dtk0, bu[2 * kb],     bv[2 * kb]);
        slab_forcing(cbase + j0 + 1, TAx, TAy, dtk0, bu[2 * kb + 1], bv[2 * kb + 1]);
    }
#pragma unroll
    for (int kb = 0; kb < 4; ++kb) {
        BP[kb] = v2f{ comp ? bv[2 * kb]     : bu[2 * kb],
                      comp ? bv[2 * kb + 1] : bu[2 * kb + 1] };
        BQ[kb] = v2f{ comp ? bu[2 * kb]     : -bv[2 * kb],
                      comp ? bu[2 * kb + 1] : -bv[2 * kb + 1] };
    }

    // -- back-to-back matrix ops: D = P@B + Q@B'
    v8f acc = {};
#pragma unroll
    for (int kb = 0; kb < 4; ++kb)
        acc = __builtin_amdgcn_wmma_f32_16x16x4_f32(false, AP[kb], false, BP[kb],
                                                    (short)0, acc, false, false);
#pragma unroll
    for (int kb = 0; kb < 4; ++kb)
        acc = __builtin_amdgcn_wmma_f32_16x16x4_f32(false, AQ[kb], false, BQ[kb],
                                                    (short)0, acc, false, false);

    // Row 15 (vgpr 7, hi lanes) = chunk totals s_q. Intra-wave chunk scan +
    // wave total via lane shuffles (convergent).
    const float L16r = s_Tre[32], L16i = s_Tim[32];
    float d7 = acc[7];
    float zcr = 0.f, zci = 0.f, wtr = 0.f, wti = 0.f;
#pragma unroll
    for (int p = 0; p < 8; ++p) {
        float sr = __shfl(d7, 16 + 2 * p);
        float si = __shfl(d7, 17 + 2 * p);
        float tr = wtr * L16r - wti * L16i + sr;   // wave total accum
        float ti = wtr * L16i + wti * L16r + si;
        wtr = tr; wti = ti;
        if (p < q) {                               // carry into chunk q
            float ur = zcr * L16r - zci * L16i + sr;
            float ui = zcr * L16i + zci * L16r + si;
            zcr = ur; zci = ui;
        }
    }
    if (lane == 0) { s_wr[w] = wtr; s_wi[w] = wti; }
    __syncthreads();

    // wave-start state: Z_blk scanned through preceding waves
    float zwr = Z[2 * blk], zwi = Z[2 * blk + 1];
    const float L128r = s_p128r, L128i = s_p128i;
#pragma unroll
    for (int p = 0; p < 7; ++p) {
        if (p < w) {
            float nr = zwr * L128r - zwi * L128i + s_wr[p];
            float ni = zwr * L128i + zwi * L128r + s_wi[p];
            zwr = nr; zwi = ni;
        }
    }
    // chunk-start state: lam^(16q)*zw + zc
    float pr = s_p16r[q], pi = s_p16i[q];
    float zqr = pr * zwr - pi * zwi + zcr;
    float zqi = pr * zwi + pi * zwr + zci;

    // out[i] = D[i] + lam^(i+1)*z0 ; lane holds rows i = r + 8*hi, column comp
    if (blk < NBLK - 1) {
#pragma unroll
        for (int r = 0; r < 8; ++r) {
            int   i   = r + 8 * hi;
            float pwr = s_Tre[17 + i], pwi = s_Tim[17 + i];
            float carry = comp ? (pwr * zqi + pwi * zqr) : (pwr * zqr - pwi * zqi);
            out[2 * (cbase + i) + comp] = acc[r] + carry;
        }
    } else {
#pragma unroll
        for (int r = 0; r < 8; ++r) {
            int   i   = r + 8 * hi;
            float pwr = s_Tre[17 + i], pwi = s_Tim[17 + i];
            float carry = comp ? (pwr * zqi + pwi * zqr) : (pwr * zqr - pwi * zqi);
            int step = cbase + i;
            if (step < NSTEPS) out[2 * step + comp] = acc[r] + carry;
        }
    }
}

// ---------------------------------------------------------------------------
extern "C" void kernel_launch(void* const* d_in, const int* in_sizes, int n_in,
                              void* d_out, int out_size, void* d_ws, size_t ws_size,
                              hipStream_t stream) {
    const float* pk  = (const float*)d_in[0];
    const float* TAx = (const float*)d_in[1];
    const float* TAy = (const float*)d_in[2];
    const float* fc  = (const float*)d_in[3];
    float* out  = (float*)d_out;
    float* Sbuf = (float*)d_ws;            // 2*NBLK floats
    float* Zbuf = Sbuf + 2 * NBLK;         // 2*NBLK floats

    k1_blocksums<<<NBLK, 256, 0, stream>>>(pk, TAx, TAy, fc, Sbuf);
    k2_scan<<<1, 32, 0, stream>>>(pk, fc, Sbuf, Zbuf);
    k3_materialize<<<NBLK, 256, 0, stream>>>(pk, TAx, TAy, fc, Zbuf, out);
}